// Model_50422916055388
// MI455X (gfx1250) — compile-verified
//
#include <hip/hip_runtime.h>
#include <hip/hip_bf16.h>

// MI455X (gfx1250) implementation of the 14-qubit VQC reference.
//
// Strategy (see analysis): one workgroup per batch element; entire state
// (2x 128KB ping-pong, SoA re/im) lives in LDS (320KB/WGP on CDNA5).
// Per layer, the 5 per-qubit rotations fuse into one 2x2 complex gate U_q;
// groups of 4 qubits kron into a 16x16 complex matrix applied with
// V_WMMA_F32_16X16X4_F32 (4 K-chunks of 4) as 4 real matmuls per tile.
// The 13-CNOT chain collapses to the permutation  dst[j] = src[j ^ (j>>1)].

#define NQ      14
#define DIM     16384            // 2^14
#define NLAYERS 2
#define BATCH   64
#define TPB     256              // 8 wave32 per workgroup

typedef float v2f __attribute__((ext_vector_type(2)));
typedef float v8f __attribute__((ext_vector_type(8)));

struct cplx { float re, im; };
__device__ __forceinline__ cplx cmul(cplx a, cplx b) {
    return { a.re * b.re - a.im * b.im, a.re * b.im + a.im * b.re };
}
__device__ __forceinline__ cplx cadd(cplx a, cplx b) {
    return { a.re + b.re, a.im + b.im };
}

struct mat2 { cplx e[2][2]; };
__device__ __forceinline__ mat2 mmul2(const mat2& X, const mat2& Y) {
    mat2 r;
#pragma unroll
    for (int i = 0; i < 2; ++i)
#pragma unroll
        for (int j = 0; j < 2; ++j)
            r.e[i][j] = cadd(cmul(X.e[i][0], Y.e[0][j]),
                             cmul(X.e[i][1], Y.e[1][j]));
    return r;
}

// D = A(16x4) * B(4x16) + C ; fp32 WMMA, wave32, EXEC must be all ones.
__device__ __forceinline__ v8f wmma4(v2f a, v2f b, v8f c) {
    return __builtin_amdgcn_wmma_f32_16x16x4_f32(
        /*neg_a=*/false, a, /*neg_b=*/false, b,
        /*c_mod=*/(short)0, c, /*reuse_a=*/false, /*reuse_b=*/false);
}

// Dynamic LDS layout (floats):
//   [0      , DIM)    stateA re
//   [DIM    , 2*DIM)  stateA im
//   [2*DIM  , 3*DIM)  stateB re
//   [3*DIM  , 4*DIM)  stateB im
//   [4*DIM  , +256)   gate re (16x16 row-major)
//   [..     , +256)   gate im
//   [..     , +64 )   per-qubit fused gate re  (14 x 2x2)
//   [..     , +64 )   per-qubit fused gate im
//   [..     , +256)   reduction scratch
#define SMEM_FLOATS (4 * DIM + 256 + 256 + 64 + 64 + 256)
#define SMEM_BYTES  (SMEM_FLOATS * 4)

__global__ void __launch_bounds__(TPB)
vqc14_kernel(const float* __restrict__ x,
             const float* __restrict__ thetas,
             float* __restrict__ out)
{
    extern __shared__ float smem[];
    float* sAre = smem;
    float* sAim = smem + DIM;
    float* sBre = smem + 2 * DIM;
    float* sBim = smem + 3 * DIM;
    float* gre  = smem + 4 * DIM;          // 256
    float* gim  = gre + 256;               // 256
    float* ure  = gim + 256;               // 14*4
    float* uim  = ure + 64;                // 14*4
    float* red  = uim + 64;                // 256

    const int tid  = threadIdx.x;
    const int lane = tid & 31;
    const int wv   = tid >> 5;
    const int b    = blockIdx.x;

    // |0...0> initial state
    for (int i = tid; i < DIM; i += TPB) {
        sAre[i] = (i == 0) ? 1.f : 0.f;
        sAim[i] = 0.f;
    }
    __syncthreads();

    float* src_re = sAre; float* src_im = sAim;
    float* dst_re = sBre; float* dst_im = sBim;

    const float xv = x[b];
    const float x1 = asinf(xv);        // RY feature angle
    const float x2 = acosf(xv * xv);   // RZ feature angle

    for (int layer = 0; layer < NLAYERS; ++layer) {
        // ---- fuse the 5 per-qubit rotations into one 2x2 complex gate ----
        if (tid < NQ) {
            const int q = tid;
            const float t0 = thetas[(layer * NQ + q) * 3 + 0];
            const float t1 = thetas[(layer * NQ + q) * 3 + 1];
            const float t2 = thetas[(layer * NQ + q) * 3 + 2];
            mat2 ry;  { float c = cosf(0.5f * x1), s = sinf(0.5f * x1);
                ry.e[0][0] = {c, 0.f}; ry.e[0][1] = {-s, 0.f};
                ry.e[1][0] = {s, 0.f}; ry.e[1][1] = { c, 0.f}; }
            mat2 rzf; { float c = cosf(0.5f * x2), s = sinf(0.5f * x2);
                rzf.e[0][0] = {c, -s}; rzf.e[0][1] = {0.f, 0.f};
                rzf.e[1][0] = {0.f, 0.f}; rzf.e[1][1] = {c, s}; }
            mat2 rx0; { float c = cosf(0.5f * t0), s = sinf(0.5f * t0);
                rx0.e[0][0] = {c, 0.f}; rx0.e[0][1] = {0.f, -s};
                rx0.e[1][0] = {0.f, -s}; rx0.e[1][1] = {c, 0.f}; }
            mat2 rz1; { float c = cosf(0.5f * t1), s = sinf(0.5f * t1);
                rz1.e[0][0] = {c, -s}; rz1.e[0][1] = {0.f, 0.f};
                rz1.e[1][0] = {0.f, 0.f}; rz1.e[1][1] = {c, s}; }
            mat2 rx2; { float c = cosf(0.5f * t2), s = sinf(0.5f * t2);
                rx2.e[0][0] = {c, 0.f}; rx2.e[0][1] = {0.f, -s};
                rx2.e[1][0] = {0.f, -s}; rx2.e[1][1] = {c, 0.f}; }
            // applied in order RY, RZ, RX(t0), RZ(t1), RX(t2)  =>  matrix product:
            mat2 U = mmul2(rx2, mmul2(rz1, mmul2(rx0, mmul2(rzf, ry))));
#pragma unroll
            for (int i = 0; i < 2; ++i)
#pragma unroll
                for (int j = 0; j < 2; ++j) {
                    ure[q * 4 + i * 2 + j] = U.e[i][j].re;
                    uim[q * 4 + i * 2 + j] = U.e[i][j].im;
                }
        }
        __syncthreads();

        // ---- 4 group sweeps: qubits 0-3, 4-7, 8-11, (I,I,12,13) ----
        for (int g = 0; g < 4; ++g) {
            // build the 16x16 Kronecker gate: one element per thread
            {
                const int m = tid >> 4, k = tid & 15;
                cplx e = {1.f, 0.f};
#pragma unroll
                for (int f = 0; f < 4; ++f) {       // f=0 -> group bit 3 (MSB)
                    const int mb = (m >> (3 - f)) & 1;
                    const int kb = (k >> (3 - f)) & 1;
                    const int q  = (g < 3) ? (4 * g + f)
                                           : ((f >= 2) ? (12 + (f - 2)) : -1);
                    cplx v;
                    if (q < 0) v = { (mb == kb) ? 1.f : 0.f, 0.f };   // identity pad
                    else       v = { ure[q * 4 + mb * 2 + kb],
                                     uim[q * 4 + mb * 2 + kb] };
                    e = cmul(e, v);
                }
                gre[tid] = e.re;
                gim[tid] = e.im;
            }
            __syncthreads();

            // index mapping: element(k, col n) = base(n) + k*stride
            const int lowbits = (g < 3) ? (10 - 4 * g) : 0;
            const int stride  = 1 << lowbits;
            const int lomask  = stride - 1;

            // A fragments (16x4 chunks, lanes 0-15: K=kb..kb+1; lanes 16-31: +2)
            const int mA = lane & 15;
            const int kb = (lane >> 4) << 1;
            v2f aR[4], aI[4];
#pragma unroll
            for (int c = 0; c < 4; ++c) {
                const int k0 = 4 * c + kb;
                aR[c].x = gre[mA * 16 + k0]; aR[c].y = gre[mA * 16 + k0 + 1];
                aI[c].x = gim[mA * 16 + k0]; aI[c].y = gim[mA * 16 + k0 + 1];
            }

            // each wave owns 8 disjoint 16x16 tiles -> in-place update is safe
            for (int tt = 0; tt < 8; ++tt) {
                const int t    = wv * 8 + tt;
                const int n    = t * 16 + (lane & 15);
                const int base = ((n >> lowbits) << (lowbits + 4)) | (n & lomask);

                v2f bR[4], bI[4];
#pragma unroll
                for (int c = 0; c < 4; ++c) {
                    const int k0 = 4 * c + kb;
                    const int a0 = base + k0 * stride;
                    const int a1 = a0 + stride;
                    bR[c].x = src_re[a0]; bR[c].y = src_re[a1];
                    bI[c].x = src_im[a0]; bI[c].y = src_im[a1];
                }

                v8f acc_rr = {0,0,0,0,0,0,0,0};
                v8f acc_ii = {0,0,0,0,0,0,0,0};
                v8f acc_ri = {0,0,0,0,0,0,0,0};
                v8f acc_ir = {0,0,0,0,0,0,0,0};
#pragma unroll
                for (int c = 0; c < 4; ++c) {
                    acc_rr = wmma4(aR[c], bR[c], acc_rr);   // Gr*Sr
                    acc_ii = wmma4(aI[c], bI[c], acc_ii);   // Gi*Si
                    acc_ri = wmma4(aR[c], bI[c], acc_ri);   // Gr*Si
                    acc_ir = wmma4(aI[c], bR[c], acc_ir);   // Gi*Sr
                }

                // D layout: lane L, VGPR v -> row v + 8*(L>>4), col L&15
#pragma unroll
                for (int v = 0; v < 8; ++v) {
                    const int m    = v + ((lane >> 4) << 3);
                    const int addr = base + m * stride;
                    src_re[addr] = acc_rr[v] - acc_ii[v];
                    src_im[addr] = acc_ri[v] + acc_ir[v];
                }
            }
            __syncthreads();
        }

        // ---- CNOT chain (q -> q+1, q=0..12) == one permutation ----
        // dst[j] = src[j ^ (j>>1)]   (inverse of MSB-first prefix-XOR scan)
        for (int j = tid; j < DIM; j += TPB) {
            const int i = j ^ (j >> 1);
            dst_re[j] = src_re[i];
            dst_im[j] = src_im[i];
        }
        __syncthreads();
        { float* t = src_re; src_re = dst_re; dst_re = t; }
        { float* t = src_im; src_im = dst_im; dst_im = t; }
    }

    // ---- <Z^n> = sum_k parity(k) * |psi_k|^2 ----
    float local = 0.f;
    for (int i = tid; i < DIM; i += TPB) {
        const float pr = src_re[i], pi = src_im[i];
        const float p  = pr * pr + pi * pi;
        local += (__builtin_popcount((unsigned)i) & 1) ? -p : p;
    }
    red[tid] = local;
    __syncthreads();
#pragma unroll
    for (int s = TPB / 2; s > 0; s >>= 1) {
        if (tid < s) red[tid] += red[tid + s];
        __syncthreads();
    }
    if (tid == 0) out[b] = red[0];
}

extern "C" void kernel_launch(void* const* d_in, const int* in_sizes, int n_in,
                              void* d_out, int out_size, void* d_ws, size_t ws_size,
                              hipStream_t stream) {
    (void)in_sizes; (void)n_in; (void)out_size; (void)d_ws; (void)ws_size;
    const float* x      = (const float*)d_in[0];   // [64]
    const float* thetas = (const float*)d_in[1];   // [2,14,3]
    float* out          = (float*)d_out;           // [64]

    // 259.5 KB dynamic LDS per workgroup (<= 320 KB/WGP on CDNA5)
    hipFuncSetAttribute((const void*)vqc14_kernel,
                        hipFuncAttributeMaxDynamicSharedMemorySize, SMEM_BYTES);

    vqc14_kernel<<<BATCH, TPB, SMEM_BYTES, stream>>>(x, thetas, out);
}